// LearnableCutMix_47012712022032
// MI455X (gfx1250) — compile-verified
//
#include <hip/hip_runtime.h>
#include <hip/hip_bf16.h>
#include <math.h>
#include <stdint.h>

// Problem geometry (compile-time constants from the reference)
#define B_  64
#define C_  3
#define H_  512
#define W_  512
#define HW_ (H_ * W_)

typedef float v4f __attribute__((ext_vector_type(4)));

// ---------------- CDNA5 async global<->LDS copy path (probe-gated) ----------------
#if __has_builtin(__builtin_amdgcn_global_load_async_to_lds_b128) && \
    __has_builtin(__builtin_amdgcn_global_store_async_from_lds_b128)
#define HAVE_ASYNC_LDS 1

// Builtin parameter type (from hipcc diagnostic):
//   '__attribute__((__vector_size__(4 * sizeof(int)))) int __device__ *'
typedef int v4i_b __attribute__((vector_size(4 * sizeof(int))));
typedef __attribute__((address_space(1))) v4i_b glb_v4i;   // prints as "__device__"
typedef __attribute__((address_space(3))) v4i_b lds_v4i;   // LDS

// AS1 global address == generic address (identity), via integer round-trip.
__device__ __forceinline__ glb_v4i* to_glb(const void* p) {
    return (glb_v4i*)(uintptr_t)p;
}
// Generic pointers into LDS carry the LDS byte offset in their low 32 bits
// (ISA flat-aperture rule), so truncation gives AS3.
__device__ __forceinline__ lds_v4i* to_lds(void* p) {
    return (lds_v4i*)(unsigned)(uintptr_t)p;
}

#if __has_builtin(__builtin_amdgcn_s_wait_asynccnt)
#define WAIT_ASYNC0() __builtin_amdgcn_s_wait_asynccnt(0)
#else
#define WAIT_ASYNC0() asm volatile("s_wait_asynccnt 0x0" ::: "memory")
#endif
#endif  // async builtins

// ---------------- small deterministic RNG (counter-hash based) ----------------
__device__ __forceinline__ unsigned hash_u32(unsigned x) {
    x ^= x >> 16; x *= 0x7feb352dU;
    x ^= x >> 15; x *= 0x846ca68bU;
    x ^= x >> 16;
    return x;
}

struct RngState { unsigned seed; unsigned ctr; };

__device__ __forceinline__ float next_uniform(RngState& r) {
    unsigned v = hash_u32(r.seed ^ (0x9e3779b9U * (++r.ctr)));
    return ((float)(v >> 8) + 0.5f) * (1.0f / 16777216.0f);  // (0,1)
}

__device__ __forceinline__ float next_normal(RngState& r) {
    float u1 = next_uniform(r);
    float u2 = next_uniform(r);
    return sqrtf(-2.0f * logf(u1)) * cosf(6.2831853071795864f * u2);
}

// Marsaglia-Tsang gamma sampler (a in [1e-4, 100] here)
__device__ float gamma_sample(float a, RngState& r) {
    float boost = 1.0f;
    if (a < 1.0f) {
        float u = next_uniform(r);
        boost = powf(u, 1.0f / a);
        a += 1.0f;
    }
    float d = a - (1.0f / 3.0f);
    float c = rsqrtf(9.0f * d);
    float res = d;  // fallback if rejection loop exhausts
    #pragma unroll 1
    for (int it = 0; it < 32; ++it) {
        float x = next_normal(r);
        float t = 1.0f + c * x;
        if (t <= 0.0f) continue;
        float v = t * t * t;
        float u = next_uniform(r);
        if (logf(u) < 0.5f * x * x + d - d * v + d * logf(v)) { res = d * v; break; }
    }
    return boost * res;
}

// ---------------- kernel 1: per-sample scalar pipeline (B=64) ----------------
__global__ __launch_bounds__(64)
void bbox_kernel(const float* __restrict__ eps,
                 const float* __restrict__ mu_p,
                 const float* __restrict__ log_sigma_p,
                 const int*   __restrict__ y,
                 const int*   __restrict__ cx,
                 const int*   __restrict__ cy,
                 const int*   __restrict__ perm,
                 int4*        __restrict__ bbox_ws,   // [B] {bbx1,bbx2,bby1,bby2}
                 float*       __restrict__ out_tail)  // [3*B]: y, y[perm], final_lambda
{
    int i = threadIdx.x;
    if (i >= B_) return;

    const float mu        = mu_p[0];
    const float log_sigma = log_sigma_p[0];

    float la = mu + expf(log_sigma) * eps[i];
    float a  = fminf(fmaxf(expf(la), 1e-4f), 100.0f);

    RngState r; r.seed = hash_u32(42u ^ (unsigned)(i * 0x85ebca6b)); r.ctr = 0;
    float g1 = gamma_sample(a, r);
    float g2 = gamma_sample(a, r);
    float lam1 = g1 / (g1 + g2);
    lam1 = fminf(fmaxf(lam1, 1e-6f), 1.0f - 1e-6f);

    float lam_patch = fminf(fmaxf(1.0f - lam1, 0.0f), 1.0f);
    float cut_rat   = sqrtf(lam_patch);
    int cut_w = (int)floorf(512.0f * cut_rat);
    int cut_h = (int)floorf(512.0f * cut_rat);

    int bbx1 = min(max(cx[i] - cut_w / 2, 0), 512);
    int bbx2 = min(max(cx[i] + cut_w / 2, 0), 512);
    int bby1 = min(max(cy[i] - cut_h / 2, 0), 512);
    int bby2 = min(max(cy[i] + cut_h / 2, 0), 512);

    bool valid = (cut_w > 0) && (cut_h > 0) && (bbx1 < bbx2) && (bby1 < bby2);
    if (!valid) { bbx1 = bbx2 = bby1 = bby2 = 0; }

    bbox_ws[i] = make_int4(bbx1, bbx2, bby1, bby2);

    float area = (float)((bbx2 - bbx1) * (bby2 - bby1));
    float fl   = valid ? (1.0f - area * (1.0f / (512.0f * 512.0f))) : 1.0f;
    fl = fminf(fmaxf(fl, 0.0f), 1.0f);

    out_tail[i]          = (float)y[i];
    out_tail[B_ + i]     = (float)y[perm[i]];
    out_tail[2 * B_ + i] = fl;
}

// ---------------- kernel 2: bandwidth-bound masked select ----------------
// grid = (256, B*C); block = 256 threads; one float4 per thread.
// A wave32 covers 128 contiguous columns of one row of one (b,c) image, so
// the row test and (except at box edges) the column test are wave-uniform.
// Pure-copy lanes (~99%) go through the CDNA5 async DMA path:
//   global_load_async_to_lds_b128 -> s_wait_asynccnt -> global_store_async_from_lds_b128
// Only bbox-column-edge lanes need a VGPR blend.
__global__ __launch_bounds__(256)
void mix_kernel(const float* __restrict__ x,
                const int*   __restrict__ perm,
                const int4*  __restrict__ bbox,
                float*       __restrict__ out)
{
#if defined(HAVE_ASYNC_LDS)
    __shared__ v4f stage[256];   // one 16B slot per thread
#endif

    const int bc = blockIdx.y;          // 0 .. B*C-1
    const int b  = bc / C_;
    const int c  = bc - b * C_;

    const int p  = blockIdx.x * 256 + threadIdx.x;  // vec4 index within the image
    const int h  = p >> 7;                           // W/4 = 128 vec4 per row
    const int w0 = (p & 127) << 2;                   // starting column of this vec4

    const int4 bb = bbox[b];            // uniform per block
    const int  pb = perm[b];            // uniform per block

    const v4f* __restrict__ x4   = (const v4f*)x;
    v4f*       __restrict__ out4 = (v4f*)out;

    const size_t didx = (size_t)bc * (HW_ / 4) + (size_t)p;             // own-image vec4 idx
    const size_t sidx = (size_t)(pb * C_ + c) * (HW_ / 4) + (size_t)p;  // shuffled-source vec4 idx

    const bool row_in = (h >= bb.z) && (h < bb.w);
    const bool any    = row_in && (w0 < bb.y) && (w0 + 4 > bb.x);
    const bool all    = row_in && (w0 >= bb.x) && (w0 + 4 <= bb.y);
    const bool blend  = any && !all;

#if defined(HAVE_ASYNC_LDS)
    if (!blend) {
        // pure 16B copy: async DMA global -> LDS -> global, no VGPR staging
        const v4f* src = all ? (x4 + sidx) : (x4 + didx);
        __builtin_amdgcn_global_load_async_to_lds_b128(
            to_glb((const void*)src), to_lds((void*)&stage[threadIdx.x]), 0, 0);
        WAIT_ASYNC0();  // LDS slot populated
        __builtin_amdgcn_global_store_async_from_lds_b128(
            to_glb((const void*)(out4 + didx)), to_lds((void*)&stage[threadIdx.x]), 0, 0);
        // store completion tracked by ASYNCcnt; S_ENDPGM performs implicit wait-idle
    } else {
        v4f base = __builtin_nontemporal_load(&x4[didx]);
        v4f sh   = __builtin_nontemporal_load(&x4[sidx]);
        v4f res;
        #pragma unroll
        for (int k = 0; k < 4; ++k) {
            int wc = w0 + k;
            res[k] = (wc >= bb.x && wc < bb.y) ? sh[k] : base[k];
        }
        __builtin_nontemporal_store(res, &out4[didx]);
    }
#else
    // Fallback: direct NT b128 path (also taken by the host compilation pass)
    __builtin_prefetch((const void*)(x4 + didx), 0, 0);
    v4f res;
    if (all) {
        res = __builtin_nontemporal_load(&x4[sidx]);
    } else if (blend) {
        v4f base = __builtin_nontemporal_load(&x4[didx]);
        v4f sh   = __builtin_nontemporal_load(&x4[sidx]);
        #pragma unroll
        for (int k = 0; k < 4; ++k) {
            int wc = w0 + k;
            res[k] = (wc >= bb.x && wc < bb.y) ? sh[k] : base[k];
        }
    } else {
        res = __builtin_nontemporal_load(&x4[didx]);
    }
    __builtin_nontemporal_store(res, &out4[didx]);
#endif
}

// ---------------- launch ----------------
extern "C" void kernel_launch(void* const* d_in, const int* in_sizes, int n_in,
                              void* d_out, int out_size, void* d_ws, size_t ws_size,
                              hipStream_t stream) {
    // setup_inputs() order: x, eps, mu, log_sigma, y, cx, cy, perm
    const float* x         = (const float*)d_in[0];
    const float* eps       = (const float*)d_in[1];
    const float* mu        = (const float*)d_in[2];
    const float* log_sigma = (const float*)d_in[3];
    const int*   y         = (const int*)d_in[4];
    const int*   cx        = (const int*)d_in[5];
    const int*   cy        = (const int*)d_in[6];
    const int*   perm      = (const int*)d_in[7];

    float* out = (float*)d_out;
    int4*  bbx = (int4*)d_ws;   // 64 * 16B = 1 KiB of scratch

    const size_t N = (size_t)B_ * C_ * H_ * W_;   // x_mixed elements

    bbox_kernel<<<dim3(1), dim3(64), 0, stream>>>(eps, mu, log_sigma, y, cx, cy, perm,
                                                  bbx, out + N);

    dim3 grid(256, B_ * C_);
    mix_kernel<<<grid, dim3(256), 0, stream>>>(x, perm, bbx, out);
}